// AttentionLinear_30709016166931
// MI455X (gfx1250) — compile-verified
//
#include <hip/hip_runtime.h>
#include <hip/hip_bf16.h>
#include <cmath>

// ---------------------------------------------------------------------------
// AttentionLinear for MI455X (gfx1250, wave32).
//  Phase 1: attn = X(16n x 512) @ blockdiag(W)(512 x 16) via V_WMMA_F32_16X16X4_F32,
//           A-tile staged Global->LDS with the Tensor Data Mover (TDM),
//           LDS rows padded (stride 516 dw) so ds_load_b64 is bank-conflict-free.
//  Phases 2-4: bandwidth-bound edge passes; segment buffers (3.2 MB) are L2-resident.
//  Roofline: ~0.49 GB total traffic -> ~21 us at 23.3 TB/s; compute is trivial.
// ---------------------------------------------------------------------------

typedef __attribute__((ext_vector_type(2))) float    v2f;
typedef __attribute__((ext_vector_type(8))) float    v8f;
typedef __attribute__((ext_vector_type(4))) unsigned u32x4;
typedef __attribute__((ext_vector_type(4))) int      i32x4;
typedef __attribute__((ext_vector_type(8))) int      i32x8;

#define TILE_ROW_DW 516                    // 512 data dwords + 2x2 pad dwords
#define TILE_BYTES  (16 * TILE_ROW_DW * 4) // 33,024 B per 16-node tile

// Order-preserving float<->uint map so segment-max can use integer atomicMax
// (deterministic: max is order-independent).
__device__ __forceinline__ unsigned f2ord(float f) {
  unsigned u = __float_as_uint(f);
  return (u & 0x80000000u) ? ~u : (u | 0x80000000u);
}
__device__ __forceinline__ float ord2f(unsigned u) {
  return __uint_as_float((u & 0x80000000u) ? (u & 0x7fffffffu) : ~u);
}

__global__ __launch_bounds__(256) void k_init(unsigned* __restrict__ mbuf,
                                              unsigned* __restrict__ sbuf, int n) {
  int i = blockIdx.x * blockDim.x + threadIdx.x;
  if (i < n) {
    const unsigned ninf = f2ord(-__builtin_inff());  // 0x007FFFFF
    mbuf[i] = ninf;
    sbuf[i] = ninf;
  }
}

// LDS offset (in dwords) of dword f within a padded row: 2 pad dwords after
// every 256 data dwords (matches TDM pad_interval=256dw / pad_amount=2dw).
__device__ __forceinline__ constexpr int pad_off(int f) {
  return f + ((f >= 256) ? 2 : 0);
}

// Phase 1: one wave32 / one 16-node tile per block.
// A (16x4 f32 per step): lane = half*16 + row holds X[row, kb], X[row, kb+1],
//   kb = kstep*4 + half*2  (ISA 16x4 f32 A layout), read from the LDS tile.
// B (4x16 f32 per step): lane = half*16 + col holds blockdiag-W[kb..kb+1, col];
//   the lane's W column is preloaded into 32 VGPRs, masked per head block with
//   branch-free selects (no divergent loads in the inner loop).
// C/D (16x16 f32): vgpr r, lanes 0-15 -> (M=r, N=lane), lanes 16-31 -> (M=8+r).
__global__ __launch_bounds__(32) void k_attn_wmma(
    const float* __restrict__ X, const float* __restrict__ Wk,
    float* __restrict__ attn, int Nn) {
  extern __shared__ char smem[];
  float* sf = (float*)smem;

  const int lane  = threadIdx.x & 31;
  const int node0 = blockIdx.x << 4;
  if (node0 >= Nn) return;
  int rows = Nn - node0;
  if (rows > 16) rows = 16;

  // ---- Stage the A tile (rows x 512 f32, contiguous in X) into LDS ----
#if __has_builtin(__builtin_amdgcn_tensor_load_to_lds)
  {
    const unsigned           ldsA = (unsigned)(unsigned long long)(uintptr_t)smem;
    const unsigned long long ga =
        (unsigned long long)(uintptr_t)(X + (size_t)node0 * 512);
    // D# group 0: count=1 valid user descriptor, lds_addr, global_addr, type=2.
    u32x4 g0 = {1u, ldsA, (unsigned)(ga & 0xffffffffu),
                (unsigned)((ga >> 32) & 0x01ffffffu) | 0x80000000u};
    // D# group 1: data_size=4B, pad_enable, pad_interval=256dw, pad_amount=2dw;
    // tensor_dim0=512, tensor_dim1=rows, tile_dim0=512, tile_dim1=rows,
    // tensor_dim0_stride=512 (contiguous rows).
    i32x8 g1;
    g1[0] = (2 << 16) | (1 << 20) | (7 << 22) | (1 << 25);
    g1[1] = (int)(512u << 16);                 // tensor_dim0[15:0] in [31:16]
    g1[2] = (int)((unsigned)rows << 16);       // tensor_dim0 hi | tensor_dim1 lo
    g1[3] = (int)(512u << 16);                 // tensor_dim1 hi | tile_dim0
    g1[4] = rows;                              // tile_dim1 | tile_dim2=0
    g1[5] = 512;                               // tensor_dim0_stride lo
    g1[6] = 0;
    g1[7] = 0;
    i32x4 g2 = {1, 0, 0, 0};                   // tensor_dim2=1 (degenerate 3rd dim)
    i32x4 g3 = {0, 0, 0, 0};
#if defined(__clang_major__) && __clang_major__ >= 23
    i32x8 g4 = {0, 0, 0, 0, 0, 0, 0, 0};
    __builtin_amdgcn_tensor_load_to_lds(g0, g1, g2, g3, g4, 0);
#else
    __builtin_amdgcn_tensor_load_to_lds(g0, g1, g2, g3, 0);
#endif
    __builtin_amdgcn_s_wait_tensorcnt((short)0);
  }
#else
  {
    const float* __restrict__ xt = X + (size_t)node0 * 512;
    for (int t = lane; t < rows * 512; t += 32) {
      int r = t >> 9, f = t & 511;
      sf[r * TILE_ROW_DW + pad_off(f)] = xt[t];
    }
    __syncthreads();
  }
#endif

  const int half = lane >> 4;   // which K-pair this lane feeds
  const int arow = lane & 15;   // A row (node within tile)
  const int col  = lane & 15;   // B / D column

  // Preload this lane's W column fragments (Wk is (D=64, H=8) row-major, 2 KB).
  float bx[16], by[16];
#pragma unroll
  for (int i = 0; i < 16; ++i) {
    const int d0 = i * 4 + half * 2;
    bx[i] = Wk[d0 * 8 + (col & 7)];
    by[i] = Wk[(d0 + 1) * 8 + (col & 7)];
  }

  const float* __restrict__ arow_p = sf + arow * TILE_ROW_DW;
  v8f acc = {};
#pragma unroll
  for (int h = 0; h < 8; ++h) {
    const bool on = (col == h);       // blockdiag mask: branch-free v_cndmask
#pragma unroll
    for (int i = 0; i < 16; ++i) {
      const int kb = h * 64 + i * 4 + half * 2;
      v2f a = *(const v2f*)(arow_p + pad_off(kb));   // ds_load_b64, conflict-free
      v2f b;
      b.x = on ? bx[i] : 0.0f;
      b.y = on ? by[i] : 0.0f;
      acc = __builtin_amdgcn_wmma_f32_16x16x4_f32(
          /*neg_a=*/false, a, /*neg_b=*/false, b,
          /*c_mod=*/(short)0, acc, /*reuse_a=*/false, /*reuse_b=*/false);
    }
  }

  if (col < 8) {
    const int rb = node0 + half * 8;
#pragma unroll
    for (int r = 0; r < 8; ++r) {
      const int n = rb + r;
      if (n < Nn) attn[(size_t)n * 8 + col] = acc[r];
    }
  }
}

// Phase 2: m[tgt,h] = segment_max(tanh(attn[src,h]))
__global__ __launch_bounds__(256) void k_seg_max(
    const float* __restrict__ attn, const int* __restrict__ src,
    const int* __restrict__ tgt, unsigned* __restrict__ mbuf, int EH) {
  int idx = blockIdx.x * blockDim.x + threadIdx.x;
  if (idx >= EH) return;
  int e = idx >> 3, h = idx & 7;
  float t = tanhf(attn[(size_t)src[e] * 8 + h]);
  atomicMax(&mbuf[(size_t)tgt[e] * 8 + h], f2ord(t));
}

// Phase 3: s[tgt,h] = segment_max(exp(tanh(attn[src,h]) - m[tgt,h]))
__global__ __launch_bounds__(256) void k_seg_max_exp(
    const float* __restrict__ attn, const int* __restrict__ src,
    const int* __restrict__ tgt, const unsigned* __restrict__ mbuf,
    unsigned* __restrict__ sbuf, int EH) {
  int idx = blockIdx.x * blockDim.x + threadIdx.x;
  if (idx >= EH) return;
  int e = idx >> 3, h = idx & 7;
  size_t ti = (size_t)tgt[e] * 8 + h;
  float t = tanhf(attn[(size_t)src[e] * 8 + h]);
  float v = expf(t - ord2f(mbuf[ti]));
  atomicMax(&sbuf[ti], f2ord(v));
}

// Phase 4: out = exp(tanh(attn[src]) - m[tgt]) / (s[tgt] + 1e-9) * drop_mask
__global__ __launch_bounds__(256) void k_finalize(
    const float* __restrict__ attn, const int* __restrict__ src,
    const int* __restrict__ tgt, const unsigned* __restrict__ mbuf,
    const unsigned* __restrict__ sbuf, const float* __restrict__ drop,
    float* __restrict__ out, int EH) {
  int idx = blockIdx.x * blockDim.x + threadIdx.x;
  if (idx >= EH) return;
  int e = idx >> 3, h = idx & 7;
  size_t ti = (size_t)tgt[e] * 8 + h;
  float t = tanhf(attn[(size_t)src[e] * 8 + h]);
  float v = expf(t - ord2f(mbuf[ti]));
  out[idx] = (v / (ord2f(sbuf[ti]) + 1e-9f)) * drop[idx];
}

extern "C" void kernel_launch(void* const* d_in, const int* in_sizes, int n_in,
                              void* d_out, int out_size, void* d_ws, size_t ws_size,
                              hipStream_t stream) {
  (void)n_in; (void)out_size; (void)ws_size;
  const float* X       = (const float*)d_in[0];  // (1, N, 8, 64) f32
  const float* Wk      = (const float*)d_in[1];  // (64, 8, 1)    f32
  const int*   targets = (const int*)d_in[2];    // (1, E)
  const int*   sources = (const int*)d_in[3];    // (1, E)
  // d_in[4] = degree (unused by reference), d_in[6] = N scalar (use in_sizes instead)
  const float* drop    = (const float*)d_in[5];  // (1, E, 8)
  float*       out     = (float*)d_out;          // (1, E, 8, 1)

  const int N = in_sizes[4];   // degree has B*N elements, B == 1
  const int E = in_sizes[2];   // targets has B*E elements

  float*    attn = (float*)d_ws;                       // N*8 f32   (3.2 MB)
  unsigned* mbuf = (unsigned*)(attn + (size_t)N * 8);  // N*8 u32
  unsigned* sbuf = mbuf + (size_t)N * 8;               // N*8 u32

  const int NH8 = N * 8;
  const int EH  = E * 8;

  k_init<<<(NH8 + 255) / 256, 256, 0, stream>>>(mbuf, sbuf, NH8);

  const int tiles = (N + 15) >> 4;  // one wave32 / 16-node tile / block
  k_attn_wmma<<<tiles, 32, TILE_BYTES, stream>>>(X, Wk, attn, N);

  k_seg_max    <<<(EH + 255) / 256, 256, 0, stream>>>(attn, sources, targets, mbuf, EH);
  k_seg_max_exp<<<(EH + 255) / 256, 256, 0, stream>>>(attn, sources, targets, mbuf, sbuf, EH);
  k_finalize   <<<(EH + 255) / 256, 256, 0, stream>>>(attn, sources, targets, mbuf, sbuf,
                                                      drop, out, EH);
}